// BaselineGNNLinkPredictor_63771674411417
// MI455X (gfx1250) — compile-verified
//
#include <hip/hip_runtime.h>

#define NN 50000
#define NE 800000
#define NP 200000

typedef float v2f __attribute__((ext_vector_type(2)));
typedef float v8f __attribute__((ext_vector_type(8)));

#define RS 68   // LDS row stride in dwords: 272B (16B aligned), banks shift 4/row
#define KC 64   // K-chunk staged per double-buffer slot

// ---------------------------------------------------------------------------
// f32 WMMA GEMM: C[M,N] = (RELU?)(A[M,K] @ W[K,N] + (BIAS?bias:0))
// Block = 8 waves; each wave owns a 16(M) x 64(N) tile -> 4 independent
// accumulators (no WMMA->WMMA RAW chain). The 64-wide W panel is staged into
// LDS in 64-row chunks with global_load_async_to_lds_b128, double buffered:
// chunk i+1 streams in (ASYNCcnt) while chunk i feeds the WMMAs from LDS.
// Requires M%16==0, N%64==0, K%64==0 (true for all four GEMMs here).
// grid = (ceil(M/16/8), N/64)
// ---------------------------------------------------------------------------
template <bool RELU, bool BIAS>
__global__ void gemm_wmma_f32(const float* __restrict__ A,
                              const float* __restrict__ W,
                              const float* __restrict__ bias,
                              float* __restrict__ C, int M, int K, int N) {
  __shared__ __align__(16) float ldsW[2][KC * RS];
  const int waveInBlk = threadIdx.x >> 5;
  const int lane = threadIdx.x & 31;
  const int lo = lane & 15, hi = lane >> 4;
  const int tilesM = M >> 4;
  const int tm = blockIdx.x * 8 + waveInBlk;
  const int n0 = blockIdx.y * 64;
  const bool active = (tm < tilesM);  // wave-uniform

  // Stage one 64-row x 64-col chunk of W into LDS (async, b128 per lane-op).
  auto stage = [&](int kc, int bufIdx) {
    const float* Wp = W + (size_t)kc * N + n0;
    unsigned ldsBase = (unsigned)(uintptr_t)(&ldsW[bufIdx][0]);
    for (int L = threadIdx.x; L < KC * 16; L += 256) {
      int kl = L >> 4;            // row in chunk
      int nb = (L & 15) << 2;     // 4-dword group in row
      unsigned laddr = ldsBase + (unsigned)(kl * RS + nb) * 4u;
      const float* gptr = Wp + (size_t)kl * N + nb;
      asm volatile("global_load_async_to_lds_b128 %0, %1, off"
                   :: "v"(laddr), "v"(gptr)
                   : "memory");
    }
  };

  stage(0, 0);

  v8f acc[4] = {};
  const float* Arow =
      active ? (A + (size_t)(tm * 16 + lo) * K + 2 * hi) : A;
  const int nChunks = K / KC;
  for (int ci = 0; ci < nChunks; ++ci) {
    asm volatile("s_wait_asynccnt 0" ::: "memory");
    __syncthreads();  // chunk ci resident & visible to all waves
    if (ci + 1 < nChunks) stage((ci + 1) * KC, (ci + 1) & 1);
    if (active) {
      const float* buf = ldsW[ci & 1];
      const int kc = ci * KC;
      for (int kl = 0; kl < KC; kl += 4) {
        v2f a = *(const v2f*)(Arow + kc + kl);  // A frag reused by 4 tiles
        int kb = kl + 2 * hi;
#pragma unroll
        for (int t = 0; t < 4; ++t) {
          v2f b;
          b.x = buf[kb * RS + t * 16 + lo];
          b.y = buf[(kb + 1) * RS + t * 16 + lo];
          acc[t] = __builtin_amdgcn_wmma_f32_16x16x4_f32(
              false, a, false, b, (short)0, acc[t], false, false);
        }
      }
    }
    __syncthreads();  // all reads of buf[ci&1] done before it is re-staged
  }
  if (!active) return;
#pragma unroll
  for (int t = 0; t < 4; ++t) {
    int cc = n0 + t * 16 + lo;
    float bv = BIAS ? bias[cc] : 0.0f;
#pragma unroll
    for (int r = 0; r < 8; ++r) {
      int rr = tm * 16 + r + 8 * hi;  // C/D layout: VGPR r -> rows r / r+8
      float v = acc[t][r] + bv;
      if (RELU) v = fmaxf(v, 0.0f);
      C[(size_t)rr * N + cc] = v;
    }
  }
}

// ---------------------------------------------------------------------------
// Link-predictor first GEMM with gathered A rows:
//   pair[p] = concat(H[src[p]], H[dst[p]])  (K=512), N=64
//   Z = relu(pair @ W + b). Same LDS async staging / 4-accumulator scheme.
// grid = (ceil(P/16/8), 1)
// ---------------------------------------------------------------------------
__global__ void lp_gemm_wmma(const float* __restrict__ H,
                             const int* __restrict__ srcI,
                             const int* __restrict__ dstI,
                             const float* __restrict__ W,
                             const float* __restrict__ bias,
                             float* __restrict__ Z, int P) {
  const int K = 512, N = 64;
  __shared__ __align__(16) float ldsW[2][KC * RS];
  const int waveInBlk = threadIdx.x >> 5;
  const int lane = threadIdx.x & 31;
  const int lo = lane & 15, hi = lane >> 4;
  const int tilesM = P >> 4;
  const int tm = blockIdx.x * 8 + waveInBlk;
  const bool active = (tm < tilesM);

  auto stage = [&](int kc, int bufIdx) {
    const float* Wp = W + (size_t)kc * N;
    unsigned ldsBase = (unsigned)(uintptr_t)(&ldsW[bufIdx][0]);
    for (int L = threadIdx.x; L < KC * 16; L += 256) {
      int kl = L >> 4;
      int nb = (L & 15) << 2;
      unsigned laddr = ldsBase + (unsigned)(kl * RS + nb) * 4u;
      const float* gptr = Wp + (size_t)kl * N + nb;
      asm volatile("global_load_async_to_lds_b128 %0, %1, off"
                   :: "v"(laddr), "v"(gptr)
                   : "memory");
    }
  };

  stage(0, 0);

  int row = active ? (tm * 16 + lo) : 0;
  const float* bs = H + (size_t)srcI[row] * 256 + 2 * hi;
  const float* bd = H + (size_t)dstI[row] * 256 + 2 * hi;

  v8f acc[4] = {};
  const int nChunks = K / KC;  // 8
  for (int ci = 0; ci < nChunks; ++ci) {
    asm volatile("s_wait_asynccnt 0" ::: "memory");
    __syncthreads();
    if (ci + 1 < nChunks) stage((ci + 1) * KC, (ci + 1) & 1);
    if (active) {
      const float* buf = ldsW[ci & 1];
      const int kc = ci * KC;
      // whole chunk lies on one side of the src/dst split (256 % KC == 0)
      const float* Ab = (kc < 256) ? (bs + kc) : (bd + (kc - 256));
      for (int kl = 0; kl < KC; kl += 4) {
        v2f a = *(const v2f*)(Ab + kl);
        int kb = kl + 2 * hi;
#pragma unroll
        for (int t = 0; t < 4; ++t) {
          v2f b;
          b.x = buf[kb * RS + t * 16 + lo];
          b.y = buf[(kb + 1) * RS + t * 16 + lo];
          acc[t] = __builtin_amdgcn_wmma_f32_16x16x4_f32(
              false, a, false, b, (short)0, acc[t], false, false);
        }
      }
    }
    __syncthreads();
  }
  if (!active) return;
#pragma unroll
  for (int t = 0; t < 4; ++t) {
    int cc = t * 16 + lo;
    float bv = bias[cc];
#pragma unroll
    for (int r = 0; r < 8; ++r) {
      int rr = tm * 16 + r + 8 * hi;
      float v = fmaxf(acc[t][r] + bv, 0.0f);
      Z[(size_t)rr * N + cc] = v;
    }
  }
}

// ---------------------------------------------------------------------------
// a_src[n,h] = sum_c H[n,h,c]*att_s[h,c] ; a_dst likewise. One wave per node.
// ---------------------------------------------------------------------------
__global__ void att_scores(const float* __restrict__ H,
                           const float* __restrict__ att_s,
                           const float* __restrict__ att_d,
                           float* __restrict__ a_src, float* __restrict__ a_dst,
                           int Nn) {
  int wave = blockIdx.x * (blockDim.x >> 5) + (threadIdx.x >> 5);
  int lane = threadIdx.x & 31;
  if (wave >= Nn) return;
  const float* hrow = H + (size_t)wave * 256 + lane * 8;
  int head = lane >> 3;
  float ss = 0.0f, dd = 0.0f;
#pragma unroll
  for (int c = 0; c < 8; ++c) {
    float hv = hrow[c];
    ss += hv * att_s[lane * 8 + c];  // [H,C] row-major == flat channel index
    dd += hv * att_d[lane * 8 + c];
  }
#pragma unroll
  for (int off = 1; off < 8; off <<= 1) {
    ss += __shfl_xor(ss, off, 32);
    dd += __shfl_xor(dd, off, 32);
  }
  if ((lane & 7) == 0) {
    a_src[wave * 4 + head] = ss;
    a_dst[wave * 4 + head] = dd;
  }
}

__global__ void fill_zero(float* __restrict__ p, long n) {
  long t = (long)blockIdx.x * blockDim.x + threadIdx.x;
  if (t < n) p[t] = 0.0f;
}

// Monotonic encoding: positive floats -> bits|0x80000000, negative -> ~bits;
// unsigned compare then matches float order. key==0 means "empty segment".
__global__ void edge_logits(const int* __restrict__ ei,
                            const float* __restrict__ asrc,
                            const float* __restrict__ adst,
                            float* __restrict__ ebuf,
                            unsigned* __restrict__ mkey, int E) {
  int t = blockIdx.x * blockDim.x + threadIdx.x;
  if (t >= E * 4) return;
  int e = t >> 2, h = t & 3;
  int j = ei[e];      // source
  int i = ei[E + e];  // target
  float v = asrc[j * 4 + h] + adst[i * 4 + h];
  v = (v > 0.0f) ? v : 0.2f * v;  // leaky_relu(0.2)
  ebuf[t] = v;
  unsigned bits = __float_as_uint(v);
  unsigned key = (bits & 0x80000000u) ? ~bits : (bits | 0x80000000u);
  atomicMax(&mkey[i * 4 + h], key);
}

__global__ void decode_max(const unsigned* __restrict__ mkey,
                           float* __restrict__ mf, int n) {
  int t = blockIdx.x * blockDim.x + threadIdx.x;
  if (t >= n) return;
  unsigned key = mkey[t];
  float v = 0.0f;  // empty segment -> where(isfinite, m, 0) -> 0
  if (key != 0u) {
    unsigned bits = (key & 0x80000000u) ? (key & 0x7fffffffu) : ~key;
    if ((bits & 0x7f800000u) != 0x7f800000u) v = __uint_as_float(bits);
  }
  mf[t] = v;
}

__global__ void edge_exp(const int* __restrict__ ei, const float* __restrict__ mf,
                         float* __restrict__ ebuf, float* __restrict__ ssum,
                         int E) {
  int t = blockIdx.x * blockDim.x + threadIdx.x;
  if (t >= E * 4) return;
  int e = t >> 2, h = t & 3;
  int i = ei[E + e];
  float ex = expf(ebuf[t] - mf[i * 4 + h]);
  ebuf[t] = ex;  // overwrite logits with exp values
  atomicAdd(&ssum[i * 4 + h], ex);
}

// One wave per edge; lane handles 8 contiguous channels (one head per lane).
__global__ void edge_aggregate(const int* __restrict__ ei,
                               const float* __restrict__ H,
                               const float* __restrict__ ebuf,
                               const float* __restrict__ ssum,
                               float* __restrict__ out, int E) {
  int wave = blockIdx.x * (blockDim.x >> 5) + (threadIdx.x >> 5);
  int lane = threadIdx.x & 31;
  if (wave >= E) return;
  int j = ei[wave], i = ei[E + wave];
  int head = lane >> 3;
  float alpha = ebuf[wave * 4 + head] / (ssum[i * 4 + head] + 1e-16f);
  const float* hj = H + (size_t)j * 256 + lane * 8;
  float* oi = out + (size_t)i * 256 + lane * 8;
#pragma unroll
  for (int c = 0; c < 8; ++c) atomicAdd(oi + c, alpha * hj[c]);
}

__global__ void add_bias(float* __restrict__ out, const float* __restrict__ bias,
                         int Nn) {
  int t = blockIdx.x * blockDim.x + threadIdx.x;
  if (t >= Nn * 256) return;
  out[t] += bias[t & 255];
}

// z2 = relu(z1 @ W2[64,32] + b2); score = z2 @ W3[32,1] + b3. One thread/pair.
__global__ void lp_tail(const float* __restrict__ Z1, const float* __restrict__ W2,
                        const float* __restrict__ b2, const float* __restrict__ W3,
                        const float* __restrict__ b3, float* __restrict__ out,
                        int P) {
  __shared__ float sW2[64 * 32];
  __shared__ float sW3[32];
  for (int t = threadIdx.x; t < 64 * 32; t += blockDim.x) sW2[t] = W2[t];
  if (threadIdx.x < 32) sW3[threadIdx.x] = W3[threadIdx.x];
  __syncthreads();
  int p = blockIdx.x * blockDim.x + threadIdx.x;
  if (p >= P) return;
  const float* z = Z1 + (size_t)p * 64;
  float zr[64];
#pragma unroll
  for (int c = 0; c < 64; ++c) zr[c] = z[c];
  float score = b3[0];
  for (int o = 0; o < 32; ++o) {
    float acc = b2[o];
#pragma unroll
    for (int c = 0; c < 64; ++c) acc += zr[c] * sW2[c * 32 + o];
    score += fmaxf(acc, 0.0f) * sW3[o];
  }
  out[p] = score;
}

// ---------------------------------------------------------------------------
extern "C" void kernel_launch(void* const* d_in, const int* in_sizes, int n_in,
                              void* d_out, int out_size, void* d_ws,
                              size_t ws_size, hipStream_t stream) {
  const float* x      = (const float*)d_in[0];
  const int*   ei     = (const int*)d_in[1];  // [2,NE]: row0=src j, row1=dst i
  const int*   psrc   = (const int*)d_in[2];
  const int*   pdst   = (const int*)d_in[3];
  const float* enc_W  = (const float*)d_in[4];
  const float* enc_b  = (const float*)d_in[5];
  const float* lin_W0 = (const float*)d_in[6];
  const float* att_s0 = (const float*)d_in[7];
  const float* att_d0 = (const float*)d_in[8];
  const float* bias0  = (const float*)d_in[9];
  const float* lin_W1 = (const float*)d_in[10];
  const float* att_s1 = (const float*)d_in[11];
  const float* att_d1 = (const float*)d_in[12];
  const float* bias1  = (const float*)d_in[13];
  const float* lp_W1  = (const float*)d_in[14];
  const float* lp_b1  = (const float*)d_in[15];
  const float* lp_W2  = (const float*)d_in[16];
  const float* lp_b2  = (const float*)d_in[17];
  const float* lp_W3  = (const float*)d_in[18];
  const float* lp_b3  = (const float*)d_in[19];
  float* scores = (float*)d_out;

  // Workspace layout (~183 MB of f32)
  float* ws   = (float*)d_ws;
  float* h0   = ws;                       // NN*64
  float* hlin = h0 + (size_t)NN * 64;     // NN*256 (reused later as Z1)
  float* agg0 = hlin + (size_t)NN * 256;  // NN*256
  float* agg1 = agg0 + (size_t)NN * 256;  // NN*256
  float* asrc = agg1 + (size_t)NN * 256;  // NN*4
  float* adst = asrc + (size_t)NN * 4;    // NN*4
  unsigned* mkey = (unsigned*)(adst + (size_t)NN * 4);  // NN*4
  float* mf   = (float*)(mkey + (size_t)NN * 4);        // NN*4
  float* ssum = mf + (size_t)NN * 4;      // NN*4
  float* ebuf = ssum + (size_t)NN * 4;    // NE*4

  const int BLK = 256;
  auto cdiv = [](long a, long b) { return (int)((a + b - 1) / b); };

  // 1) Node encoder: h0 = relu(x @ enc_W + enc_b)   [NN,256]x[256,64]
  {
    dim3 grid(cdiv(NN / 16, 8), 64 / 64);
    gemm_wmma_f32<true, true><<<grid, BLK, 0, stream>>>(x, enc_W, enc_b, h0,
                                                        NN, 256, 64);
  }

  // 2) Two GAT layers
  const float* layer_in[2] = {h0, agg0};
  const int    layer_K[2]  = {64, 256};
  const float* linW[2]  = {lin_W0, lin_W1};
  const float* attS[2]  = {att_s0, att_s1};
  const float* attD[2]  = {att_d0, att_d1};
  const float* biasL[2] = {bias0, bias1};
  float*       aggL[2]  = {agg0, agg1};
  for (int L = 0; L < 2; ++L) {
    dim3 grid(cdiv(NN / 16, 8), 256 / 64);
    gemm_wmma_f32<false, false><<<grid, BLK, 0, stream>>>(
        layer_in[L], linW[L], nullptr, hlin, NN, layer_K[L], 256);
    att_scores<<<cdiv(NN, 8), BLK, 0, stream>>>(hlin, attS[L], attD[L], asrc,
                                                adst, NN);
    fill_zero<<<cdiv((long)NN * 4, BLK), BLK, 0, stream>>>((float*)mkey,
                                                           (long)NN * 4);
    fill_zero<<<cdiv((long)NN * 4, BLK), BLK, 0, stream>>>(ssum, (long)NN * 4);
    fill_zero<<<cdiv((long)NN * 256, BLK), BLK, 0, stream>>>(aggL[L],
                                                             (long)NN * 256);
    edge_logits<<<cdiv((long)NE * 4, BLK), BLK, 0, stream>>>(ei, asrc, adst,
                                                             ebuf, mkey, NE);
    decode_max<<<cdiv((long)NN * 4, BLK), BLK, 0, stream>>>(mkey, mf, NN * 4);
    edge_exp<<<cdiv((long)NE * 4, BLK), BLK, 0, stream>>>(ei, mf, ebuf, ssum,
                                                          NE);
    edge_aggregate<<<cdiv(NE, 8), BLK, 0, stream>>>(ei, hlin, ebuf, ssum,
                                                    aggL[L], NE);
    add_bias<<<cdiv((long)NN * 256, BLK), BLK, 0, stream>>>(aggL[L], biasL[L],
                                                            NN);
  }

  // 3) Link predictor
  {
    float* z1 = hlin;  // reuse: NP*64 floats == NN*256 floats
    dim3 grid(cdiv(NP / 16, 8), 1);
    lp_gemm_wmma<<<grid, BLK, 0, stream>>>(agg1, psrc, pdst, lp_W1, lp_b1, z1,
                                           NP);
    lp_tail<<<cdiv(NP, BLK), BLK, 0, stream>>>(z1, lp_W2, lp_b2, lp_W3, lp_b3,
                                               scores, NP);
  }
}